// GRU_23527830847517
// MI455X (gfx1250) — compile-verified
//
#include <hip/hip_runtime.h>
#include <hip/hip_bf16.h>

// Problem constants (from reference)
#define T_STEPS 1024
#define BATCH   128
#define FDIM    243
#define HDIM    20
#define GDIM    60          // 3*H
#define POOLK   4
#define BP      512         // BATCH*POOLK
#define KPAD    244         // FDIM padded up to multiple of 4 (zero pad)
#define NROWS   (T_STEPS * BATCH)   // 131072 (t,b) rows
#define TILE_BYTES (16 * POOLK * FDIM * 4)   // feats bytes per 16-row m-tile

typedef float v2f __attribute__((ext_vector_type(2)));
typedef float v8f __attribute__((ext_vector_type(8)));

// ---------------------------------------------------------------------------
// Phase 1: maxpool(feats) -> h_seq, and x_gates = h_seq @ W_ih^T + b_ih
// GEMM M=131072, N=60 (pad 64), K=243 (pad 244), via V_WMMA_F32_16X16X4_F32.
// 128 threads (4 waves); each wave owns one 16-wide N tile; 8 M-tiles per WG.
// Next tile's feats region is prefetched (global_prefetch_b8) during WMMA.
// ---------------------------------------------------------------------------
__global__ __launch_bounds__(128)
void k_pool_xgates(const float* __restrict__ feats,
                   const float* __restrict__ W_ih,
                   const float* __restrict__ b_ih,
                   float* __restrict__ h_seq,
                   float* __restrict__ x_gates)
{
    __shared__ float Wl[64 * KPAD];   // B operand: W_ih padded [64][244]
    __shared__ float Hl[16 * KPAD];   // A operand: pooled tile [16][244]

    const int tid = threadIdx.x;

    // Stage W_ih (60x243) into zero-padded LDS
    for (int i = tid; i < 64 * KPAD; i += 128) Wl[i] = 0.0f;
    __syncthreads();
    for (int i = tid; i < GDIM * FDIM; i += 128) {
        int g = i / FDIM, f = i - g * FDIM;
        Wl[g * KPAD + f] = W_ih[i];
    }
    __syncthreads();

    const int lane = tid & 31;
    const int wv   = tid >> 5;     // n-tile 0..3
    const int lr   = lane & 15;
    const int lh   = lane >> 4;
    const int g    = wv * 16 + lr;
    const float bias = (g < GDIM) ? b_ih[g] : 0.0f;   // hoisted out of m loop

    for (int m = 0; m < 8; ++m) {
        const int rowbase = blockIdx.x * 128 + m * 16;

        // Stage pooled 16x243 tile (max over POOL=4) and persist h_seq
        for (int i = tid; i < 16 * FDIM; i += 128) {
            int rr = i / FDIM, f = i - rr * FDIM;
            int grow = rowbase + rr;
            int t = grow / BATCH, b = grow - t * BATCH;
            const float* p = feats + ((size_t)t * BP + (size_t)b * POOLK) * FDIM + f;
            float v = p[0];
            v = fmaxf(v, p[FDIM]);
            v = fmaxf(v, p[2 * FDIM]);
            v = fmaxf(v, p[3 * FDIM]);
            Hl[rr * KPAD + f] = v;
            h_seq[(size_t)grow * FDIM + f] = v;
        }
        if (tid < 16) Hl[tid * KPAD + FDIM] = 0.0f;   // zero K=243 pad column
        __syncthreads();

        // Prefetch next m-tile's contiguous 62KB feats region while WMMAs run
        if (m < 7) {
            int grow2 = blockIdx.x * 128 + (m + 1) * 16;
            int t2 = grow2 / BATCH, b2 = grow2 - t2 * BATCH;
            const char* pb =
                (const char*)(feats + ((size_t)t2 * BP + (size_t)b2 * POOLK) * FDIM);
            #pragma unroll
            for (int j = 0; j < 4; ++j) {
                size_t off = ((size_t)tid * 4 + (size_t)j) * 128;
                if (off < (size_t)TILE_BYTES) __builtin_prefetch(pb + off, 0, 1);
            }
        }

        // K loop: 61 steps of 16x16x4 fp32 WMMA
        v8f acc = {};
        const int arow = lr * KPAD;
        const int brow = (wv * 16 + lr) * KPAD;
        for (int ks = 0; ks < KPAD / 4; ++ks) {
            int k0 = ks * 4 + 2 * lh;
            v2f a, bm;
            a.x  = Hl[arow + k0];  a.y  = Hl[arow + k0 + 1];
            bm.x = Wl[brow + k0];  bm.y = Wl[brow + k0 + 1];
            acc = __builtin_amdgcn_wmma_f32_16x16x4_f32(
                false, a, false, bm, (short)0, acc, false, false);
        }

        // Add bias, store valid 60 columns
        if (g < GDIM) {
            #pragma unroll
            for (int j = 0; j < 8; ++j) {
                int row = rowbase + j + 8 * lh;
                x_gates[(size_t)row * GDIM + g] = acc[j] + bias;
            }
        }
        __syncthreads();
    }
}

// ---------------------------------------------------------------------------
// Phase 2: sequential GRU scan. One SINGLE-WAVE workgroup per batch row
// (128 WGs x 32 threads). For single-wave workgroups, barrier ops are S_NOP
// (CDNA5 ISA), so the 2 per-step syncs cost ~nothing. Each lane
// owns gates l and l+32 (covers all 60); e lives in LDS; next-step x_gates
// row is prefetched across the barrier (x_gates is L2-resident: 31.5MB).
// ---------------------------------------------------------------------------
__global__ __launch_bounds__(32)
void k_gru_scan(const float* __restrict__ e0,
                const float* __restrict__ W_hh,
                const float* __restrict__ b_hh,
                const float* __restrict__ x_gates,
                float* __restrict__ e_states)
{
    __shared__ float e[HDIM];
    __shared__ float Xs[GDIM];
    __shared__ float Hs[GDIM];

    const int b  = blockIdx.x;
    const int l  = threadIdx.x;      // 0..31
    const int g0 = l;                // < 60 always
    const int g1 = l + 32;           // valid if < 60
    const bool v1 = (g1 < GDIM);

    float whh0[HDIM], whh1[HDIM];
    const float bias0 = b_hh[g0];
    const float bias1 = v1 ? b_hh[g1] : 0.0f;
    #pragma unroll
    for (int k = 0; k < HDIM; ++k) {
        whh0[k] = W_hh[g0 * HDIM + k];
        whh1[k] = v1 ? W_hh[g1 * HDIM + k] : 0.0f;
    }
    if (l < HDIM) e[l] = e0[b * HDIM + l];
    __syncthreads();

    float xg0 = x_gates[(size_t)b * GDIM + g0];
    float xg1 = v1 ? x_gates[(size_t)b * GDIM + g1] : 0.0f;

    for (int t = 0; t < T_STEPS; ++t) {
        // two independent 20-FMA chains -> good ILP on the critical path
        float hg0 = bias0, hg1 = bias1;
        #pragma unroll
        for (int k = 0; k < HDIM; ++k) {
            float ek = e[k];
            hg0 = fmaf(whh0[k], ek, hg0);
            hg1 = fmaf(whh1[k], ek, hg1);
        }
        Xs[g0] = xg0;  Hs[g0] = hg0;
        if (v1) { Xs[g1] = xg1;  Hs[g1] = hg1; }

        // prefetch next step's x_gates row (independent of the recurrence)
        float nxg0 = 0.0f, nxg1 = 0.0f;
        if (t + 1 < T_STEPS) {
            size_t rbase = ((size_t)(t + 1) * BATCH + b) * GDIM;
            nxg0 = x_gates[rbase + g0];
            if (v1) nxg1 = x_gates[rbase + g1];
        }
        __syncthreads();   // S_NOP-class: single-wave workgroup

        if (l < HDIM) {
            float r  = 1.0f / (1.0f + __expf(-(Xs[l] + Hs[l])));
            float z  = 1.0f / (1.0f + __expf(-(Xs[l + HDIM] + Hs[l + HDIM])));
            float n  = tanhf(Xs[l + 2 * HDIM] + r * Hs[l + 2 * HDIM]);
            float en = (1.0f - z) * n + z * e[l];
            e[l] = en;
            e_states[((size_t)t * BATCH + b) * HDIM + l] = en;
        }
        __syncthreads();   // S_NOP-class: single-wave workgroup
        xg0 = nxg0;
        xg1 = nxg1;
    }
}

// ---------------------------------------------------------------------------
// Phase 3: fo = relu(e_t @ W_dec^T + b_dec); accumulate sum((fo - h_seq)^2).
// GEMM M=131072, N=243 (16 n-tiles, pad 256), K=20 via WMMA f32 16x16x4.
// 256 threads = 8 waves; 2 n-tiles per wave; one 16-row M tile per block.
// Scaled block partial -> atomicAdd into the scalar output (global mean).
// ---------------------------------------------------------------------------
__global__ __launch_bounds__(256)
void k_decode_loss(const float* __restrict__ e_states,
                   const float* __restrict__ W_dec,
                   const float* __restrict__ b_dec,
                   const float* __restrict__ h_seq,
                   float* __restrict__ out)
{
    __shared__ float Wd[256 * HDIM];   // B operand: W_dec padded [256][20]
    __shared__ float El[16 * HDIM];    // A operand: e tile [16][20]
    __shared__ float red[256];

    const int tid = threadIdx.x;
    const int rowbase = blockIdx.x * 16;

    // zero + fill hit the same thread per index (both stride 256) -> no barrier
    for (int i = tid; i < 256 * HDIM; i += 256) Wd[i] = 0.0f;
    for (int i = tid; i < FDIM * HDIM; i += 256) Wd[i] = W_dec[i];
    for (int i = tid; i < 16 * HDIM; i += 256)
        El[i] = e_states[(size_t)rowbase * HDIM + i];
    __syncthreads();

    const int lane = tid & 31;
    const int wv   = tid >> 5;      // 0..7
    const int lr   = lane & 15;
    const int lh   = lane >> 4;

    float lsum = 0.0f;
    for (int nt = wv * 2; nt < wv * 2 + 2; ++nt) {
        v8f acc = {};
        const int arow = lr * HDIM;
        const int brow = (nt * 16 + lr) * HDIM;
        #pragma unroll
        for (int ks = 0; ks < HDIM / 4; ++ks) {
            int k0 = ks * 4 + 2 * lh;
            v2f a, bm;
            a.x  = El[arow + k0];  a.y  = El[arow + k0 + 1];
            bm.x = Wd[brow + k0];  bm.y = Wd[brow + k0 + 1];
            acc = __builtin_amdgcn_wmma_f32_16x16x4_f32(
                false, a, false, bm, (short)0, acc, false, false);
        }
        const int n = nt * 16 + lr;
        if (n < FDIM) {
            float bd = b_dec[n];
            #pragma unroll
            for (int j = 0; j < 8; ++j) {
                int row = rowbase + j + 8 * lh;
                float fo = acc[j] + bd;
                fo = fo > 0.0f ? fo : 0.0f;
                float d = fo - h_seq[(size_t)row * FDIM + n];
                lsum = fmaf(d, d, lsum);
            }
        }
    }

    red[tid] = lsum;
    __syncthreads();
    for (int s = 128; s > 0; s >>= 1) {
        if (tid < s) red[tid] += red[tid + s];
        __syncthreads();
    }
    if (tid == 0)
        atomicAdd(out, red[0] * (1.0f / ((float)T_STEPS * (float)BATCH * (float)FDIM)));
}

// ---------------------------------------------------------------------------
extern "C" void kernel_launch(void* const* d_in, const int* in_sizes, int n_in,
                              void* d_out, int out_size, void* d_ws, size_t ws_size,
                              hipStream_t stream)
{
    const float* feats = (const float*)d_in[0];
    const float* e0    = (const float*)d_in[1];
    const float* W_ih  = (const float*)d_in[2];
    const float* W_hh  = (const float*)d_in[3];
    const float* b_ih  = (const float*)d_in[4];
    const float* b_hh  = (const float*)d_in[5];
    const float* W_dec = (const float*)d_in[6];
    const float* b_dec = (const float*)d_in[7];
    float* out = (float*)d_out;

    // Workspace: h_seq (127.4MB) | x_gates (31.5MB) | e_states (10.5MB)
    float* ws       = (float*)d_ws;
    float* h_seq    = ws;
    float* x_gates  = h_seq   + (size_t)NROWS * FDIM;
    float* e_states = x_gates + (size_t)NROWS * GDIM;

    hipMemsetAsync(d_out, 0, sizeof(float), stream);

    k_pool_xgates<<<NROWS / 128, 128, 0, stream>>>(feats, W_ih, b_ih, h_seq, x_gates);
    k_gru_scan<<<BATCH, 32, 0, stream>>>(e0, W_hh, b_hh, x_gates, e_states);
    k_decode_loss<<<NROWS / 16, 256, 0, stream>>>(e_states, W_dec, b_dec, h_seq, out);
}